// GetLocalFeature_5935644803390
// MI455X (gfx1250) — compile-verified
//
#include <hip/hip_runtime.h>

#define LSZ   120
#define HGT   480
#define WID   480
#define BATCH 32
#define CHN   32

typedef __attribute__((ext_vector_type(2))) float v2f;
typedef __attribute__((ext_vector_type(8))) float v8f;

// ---------------------------------------------------------------------------
// Kernel 1: per-batch agent centroid from channel 1, emit crop params to ws.
// ---------------------------------------------------------------------------
__global__ __launch_bounds__(256) void agent_pos_kernel(const float* __restrict__ img,
                                                        int* __restrict__ params)
{
    const int b = blockIdx.x;
    const float* p = img + ((size_t)b * CHN + 1) * (size_t)(HGT * WID);

    float cnt = 0.0f, sy = 0.0f, sx = 0.0f;
    for (int idx = threadIdx.x; idx < HGT * WID; idx += blockDim.x) {
        float v = p[idx];
        if (v == 1.0f) {
            cnt += 1.0f;
            sy  += (float)(idx / WID);
            sx  += (float)(idx % WID);
        }
    }

    __shared__ float rc[256], rys[256], rxs[256];
    rc[threadIdx.x] = cnt; rys[threadIdx.x] = sy; rxs[threadIdx.x] = sx;
    __syncthreads();
    for (int off = 128; off > 0; off >>= 1) {
        if ((int)threadIdx.x < off) {
            rc[threadIdx.x]  += rc[threadIdx.x + off];
            rys[threadIdx.x] += rys[threadIdx.x + off];
            rxs[threadIdx.x] += rxs[threadIdx.x + off];
        }
        __syncthreads();
    }

    if (threadIdx.x == 0) {
        float c = rc[0];
        float denom = fmaxf(c, 1.0f);
        int py = (c > 0.0f) ? (int)rintf(rys[0] / denom) : (HGT / 2);  // rintf == round-half-even
        int px = (c > 0.0f) ? (int)rintf(rxs[0] / denom) : (WID / 2);
        int ys = max(py - LSZ / 2, 0);
        int ye = min(py + LSZ / 2, HGT);
        int xs = max(px - LSZ / 2, 0);
        int xe = min(px + LSZ / 2, WID);
        params[b * 4 + 0] = ys;
        params[b * 4 + 1] = xs;
        params[b * 4 + 2] = ye - ys;   // ry
        params[b * 4 + 3] = xe - xs;   // rx
    }
}

// ---------------------------------------------------------------------------
// Kernel 2: per-(b,c) crop + separable bilinear resize.
//   Horizontal pass:  H = V(ry x rx) * Wx^T(rx x 120)   via v_wmma_f32_16x16x4_f32
//   Vertical pass:    out[i,j] = (1-wy[i])*H[y0[i],j] + wy[i]*H[y1[i],j]
// ---------------------------------------------------------------------------
__global__ __launch_bounds__(256) void crop_resize_wmma_kernel(const float* __restrict__ img,
                                                               const int* __restrict__ params,
                                                               float* __restrict__ out)
{
    __shared__ float Hbuf[128 * LSZ];          // horizontal result, rows = crop rows (padded to 128)
    __shared__ int   sx0[LSZ], sx1[LSZ], sy0[LSZ];
    __shared__ float swx[LSZ], swy[LSZ];
    __shared__ int   sprm[4];

    const int bc = blockIdx.x;
    const int b  = bc / CHN;
    const int c  = bc % CHN;

    if (threadIdx.x < 4) sprm[threadIdx.x] = params[b * 4 + threadIdx.x];
    __syncthreads();
    const int ystart = sprm[0], xstart = sprm[1], ry = sprm[2], rx = sprm[3];

    // Reference-exact sampling coordinates / weights (fp32 op-for-op).
    if (threadIdx.x < LSZ) {
        const int i = threadIdx.x;
        const float fi = (float)i + 0.5f;
        float s = fmaxf(fi * ((float)ry / (float)LSZ) - 0.5f, 0.0f);
        int y0 = (int)floorf(s);
        sy0[i] = y0;
        swy[i] = s - (float)y0;
        s = fmaxf(fi * ((float)rx / (float)LSZ) - 0.5f, 0.0f);
        int x0 = (int)floorf(s);
        sx0[i] = x0;
        sx1[i] = min(x0 + 1, rx - 1);
        swx[i] = s - (float)x0;
    }
    __syncthreads();

    const int wave    = threadIdx.x >> 5;   // 8 waves, one M-tile (16 crop rows) each
    const int lane    = threadIdx.x & 31;
    const int halfsel = lane >> 4;          // 0 -> lanes 0-15 (K=0,1), 1 -> lanes 16-31 (K=2,3)
    const int l15     = lane & 15;

    const int mTiles = (ry + 15) >> 4;
    const float* base = img + ((size_t)b * CHN + c) * (size_t)(HGT * WID);

    if (wave < mTiles) {                    // wave-uniform: EXEC all-ones inside (WMMA requirement)
        const int  m   = (wave << 4) + l15; // crop row this lane supplies in the A-fragment
        const bool mok = (m < ry);
        const float* rowptr = base + (size_t)(ystart + m) * WID + xstart;

        // Per-j-tile interpolation params for this lane's output column.
        float w0[8], w1[8];
        int   c0[8], c1[8];
        #pragma unroll
        for (int jt = 0; jt < 8; ++jt) {
            const int j = (jt << 4) + l15;
            if (j < LSZ) {
                c0[jt] = sx0[j]; c1[jt] = sx1[j];
                const float wxj = swx[j];
                w0[jt] = 1.0f - wxj; w1[jt] = wxj;
            } else {
                c0[jt] = -1; c1[jt] = -1; w0[jt] = 0.0f; w1[jt] = 0.0f;
            }
        }

        v8f acc[8];
        const v8f zero = {0.f, 0.f, 0.f, 0.f, 0.f, 0.f, 0.f, 0.f};
        #pragma unroll
        for (int jt = 0; jt < 8; ++jt) acc[jt] = zero;

        const int kIters = (rx + 3) >> 2;
        for (int kb = 0; kb < kIters; ++kb) {
            const int kk = (kb << 2) + (halfsel << 1);   // this lane supplies K = kk, kk+1
            v2f a;
            a.x = (mok && (kk     < rx)) ? rowptr[kk]     : 0.0f;
            a.y = (mok && (kk + 1 < rx)) ? rowptr[kk + 1] : 0.0f;

            #pragma unroll
            for (int jt = 0; jt < 8; ++jt) {
                v2f bf;  // Wx^T[kk..kk+1, j]: <=2 nonzeros per output column, built in VALU
                bf.x = ((kk     == c0[jt]) ? w0[jt] : 0.0f) + ((kk     == c1[jt]) ? w1[jt] : 0.0f);
                bf.y = ((kk + 1 == c0[jt]) ? w0[jt] : 0.0f) + ((kk + 1 == c1[jt]) ? w1[jt] : 0.0f);
                acc[jt] = __builtin_amdgcn_wmma_f32_16x16x4_f32(
                    false, a, false, bf, (short)0, acc[jt], false, false);
            }
        }

        // C/D layout: element g of acc[jt] -> H[wave*16 + g + 8*halfsel][jt*16 + l15]
        #pragma unroll
        for (int jt = 0; jt < 8; ++jt) {
            const int j = (jt << 4) + l15;
            if (j < LSZ) {
                #pragma unroll
                for (int g = 0; g < 8; ++g) {
                    const int row = (wave << 4) + g + (halfsel << 3);
                    Hbuf[row * LSZ + j] = acc[jt][g];
                }
            }
        }
    }
    __syncthreads();

    // Vertical lerp + coalesced store.
    float* obase = out + (size_t)bc * (LSZ * LSZ);
    for (int idx = threadIdx.x; idx < LSZ * LSZ; idx += blockDim.x) {
        const int i = idx / LSZ;
        const int j = idx % LSZ;
        const int y0 = sy0[i];
        const int y1 = min(y0 + 1, ry - 1);
        const float wy = swy[i];
        obase[idx] = Hbuf[y0 * LSZ + j] * (1.0f - wy) + Hbuf[y1 * LSZ + j] * wy;
    }
}

// ---------------------------------------------------------------------------
extern "C" void kernel_launch(void* const* d_in, const int* in_sizes, int n_in,
                              void* d_out, int out_size, void* d_ws, size_t ws_size,
                              hipStream_t stream)
{
    const float* img = (const float*)d_in[0];
    float* out = (float*)d_out;
    int* params = (int*)d_ws;   // 32 * 4 ints

    agent_pos_kernel<<<BATCH, 256, 0, stream>>>(img, params);
    crop_resize_wmma_kernel<<<BATCH * CHN, 256, 0, stream>>>(img, params, out);
}